// KeypointFlowLoss_28836410425930
// MI455X (gfx1250) — compile-verified
//
#include <hip/hip_runtime.h>
#include <math.h>

// Problem constants from the reference.
#define BATCH 16
#define KPTS  17
#define HGT   512
#define WID   512
#define BK    (BATCH * KPTS)   // 272 keypoint slots

typedef __attribute__((ext_vector_type(2))) float v2f;
typedef __attribute__((ext_vector_type(8))) float v8f;

// Single-workgroup sparse keypoint-flow loss.
//   - Phase 1: build keypoint table (pixel, disp) in LDS.
//   - Phase 2: dedup (last-write-wins) + mask (||disp|| > 0) + count.
//   - Phase 3: each masked thread gathers its pixel from all 5 flows
//              (10 independent global loads -> high MLP) and accumulates
//              gamma^(4-i)-weighted L2 norms into a private partial.
//   - Phase 4: wave32 reduction via V_WMMA_F32_16X16X4_F32 against an
//              all-ones B matrix, then cross-wave combine in LDS.
__global__ __launch_bounds__(512) void kp_flow_loss_kernel(
    const float* __restrict__ f0, const float* __restrict__ f1,
    const float* __restrict__ f2, const float* __restrict__ f3,
    const float* __restrict__ f4, const int* __restrict__ kps,
    float* __restrict__ out)
{
    __shared__ int   s_px[BK];
    __shared__ int   s_py[BK];
    __shared__ float s_dx[BK];
    __shared__ float s_dy[BK];
    __shared__ int   s_cnt;
    __shared__ float s_wsum[16];

    const int t = threadIdx.x;
    if (t == 0) s_cnt = 0;

    // ---- Phase 1: load keypoints, validity, displacement ----
    if (t < BK) {
        const int b = t / KPTS, k = t % KPTS;
        // kps layout: (B, 2, K, 2), last dim = (x, y)
        const int i0 = ((b * 2 + 0) * KPTS + k) * 2;
        const int i1 = ((b * 2 + 1) * KPTS + k) * 2;
        const int x0 = kps[i0 + 0], y0 = kps[i0 + 1];
        const int x1 = kps[i1 + 0], y1 = kps[i1 + 1];
        const bool valid = (x0 >= 0) & (y0 >= 0) & (x1 >= 0) & (y1 >= 0) &
                           (x0 < WID) & (y0 < HGT) & (x1 < WID) & (y1 < HGT);
        s_px[t] = valid ? x0 : -1;
        s_py[t] = valid ? y0 : -1;
        s_dx[t] = (float)(x1 - x0);
        s_dy[t] = (float)(y1 - y0);
    }
    __syncthreads();

    // ---- Phase 2: dedup + mask + count;  Phase 3: gather + accumulate ----
    float partial = 0.0f;
    if (t < BK) {
        const int b  = t / KPTS, k = t % KPTS;
        const int px = s_px[t];
        const int py = s_py[t];
        bool masked = false;
        float dx = s_dx[t], dy = s_dy[t];
        if (px >= 0) {
            // last-write-wins: a later keypoint of the same batch hitting the
            // same pixel overwrites this one in the reference scatter.
            bool overwritten = false;
            for (int k2 = k + 1; k2 < KPTS; ++k2) {
                const int j = b * KPTS + k2;
                overwritten |= (s_px[j] == px) & (s_py[j] == py);
            }
            masked = (!overwritten) & ((dx != 0.0f) | (dy != 0.0f));
        }
        if (masked) {
            atomicAdd(&s_cnt, 1);
            const size_t chan = (size_t)HGT * WID;
            const size_t base = (((size_t)b * 2) * HGT + (size_t)py) * WID + (size_t)px;
            // Issue all 10 gathers up front (independent loads).
            const float ax0 = f0[base], ay0 = f0[base + chan];
            const float ax1 = f1[base], ay1 = f1[base + chan];
            const float ax2 = f2[base], ay2 = f2[base + chan];
            const float ax3 = f3[base], ay3 = f3[base + chan];
            const float ax4 = f4[base], ay4 = f4[base + chan];
            const float dx0 = ax0 - dx, dy0 = ay0 - dy;
            const float dx1 = ax1 - dx, dy1 = ay1 - dy;
            const float dx2 = ax2 - dx, dy2 = ay2 - dy;
            const float dx3 = ax3 - dx, dy3 = ay3 - dy;
            const float dx4 = ax4 - dx, dy4 = ay4 - dy;
            // weights: gamma^(4-i), gamma = 0.8
            partial  = 0.4096f * sqrtf(dx0 * dx0 + dy0 * dy0);
            partial += 0.512f  * sqrtf(dx1 * dx1 + dy1 * dy1);
            partial += 0.64f   * sqrtf(dx2 * dx2 + dy2 * dy2);
            partial += 0.8f    * sqrtf(dx3 * dx3 + dy3 * dy3);
            partial += 1.0f    * sqrtf(dx4 * dx4 + dy4 * dy4);
        }
    }
    // Control flow re-converges here: EXEC is all-ones for the WMMA below.

    // ---- Phase 4: wave32 sum via WMMA f32 16x16x4 against ones ----
    // A[m][kcol]: lane (m, m+16) supply columns {0,1} / {2,3}; put the
    // partial in the first A register, zero in the second.
    // B = all ones -> D[m][n] = p[m] + p[m+16] (independent of n).
    v2f amat; amat.x = partial; amat.y = 0.0f;
    v2f bmat; bmat.x = 1.0f;    bmat.y = 1.0f;
    v8f cmat = {};
    v8f dmat = __builtin_amdgcn_wmma_f32_16x16x4_f32(
        /*neg_a=*/false, amat, /*neg_b=*/false, bmat,
        /*c_mod=*/(short)0, cmat, /*reuse_a=*/false, /*reuse_b=*/false);

    // Each lane holds 8 row-sums of D; lanes 0-15 cover m=0..7, lanes 16-31
    // cover m=8..15. Sum locally, then fold the two lane halves.
    float s = dmat[0] + dmat[1] + dmat[2] + dmat[3] +
              dmat[4] + dmat[5] + dmat[6] + dmat[7];
    s += __shfl_xor(s, 16, 32);   // full 32-lane sum in every lane

    const int lane = t & 31;
    const int wave = t >> 5;
    if (lane == 0) s_wsum[wave] = s;
    __syncthreads();

    if (t == 0) {
        float tot = 0.0f;
        #pragma unroll
        for (int w = 0; w < 16; ++w) tot += s_wsum[w];
        out[0] = tot / (float)s_cnt;   // LOSS_WEIGHT == 1.0
    }
}

extern "C" void kernel_launch(void* const* d_in, const int* in_sizes, int n_in,
                              void* d_out, int out_size, void* d_ws, size_t ws_size,
                              hipStream_t stream) {
    const float* f0  = (const float*)d_in[0];
    const float* f1  = (const float*)d_in[1];
    const float* f2  = (const float*)d_in[2];
    const float* f3  = (const float*)d_in[3];
    const float* f4  = (const float*)d_in[4];
    const int*   kps = (const int*)d_in[5];
    float*       out = (float*)d_out;
    (void)in_sizes; (void)n_in; (void)out_size; (void)d_ws; (void)ws_size;

    kp_flow_loss_kernel<<<1, 512, 0, stream>>>(f0, f1, f2, f3, f4, kps, out);
}